// GraphEncoder_39822936768787
// MI455X (gfx1250) — compile-verified
//
#include <hip/hip_runtime.h>
#include <math.h>

#define NN      32768
#define CC      128
#define EE      262144
#define RR      8
#define BB      32
#define NPGV    1024
#define KPOOLV  512
#define NSEEDS  4
#define NHEADS  4
#define DH      32
#define EPSV    1e-5f

typedef float v2f __attribute__((ext_vector_type(2)));
typedef float v8f __attribute__((ext_vector_type(8)));

// ---------------- float <-> orderable uint for atomic segment-max ------------
__device__ __forceinline__ unsigned enc_f(float f) {
    unsigned u = __float_as_uint(f);
    return (u & 0x80000000u) ? ~u : (u | 0x80000000u);
}
__device__ __forceinline__ float dec_f(unsigned u) {
    return (u & 0x80000000u) ? __uint_as_float(u & 0x7fffffffu) : __uint_as_float(~u);
}

// ---------------- fp32 WMMA GEMM, 128x64 block tile, LDS-staged B ------------
// A: Mg x Kg row-major (lda = Kg); B: Kg x Ng row-major (ldb = Ng); C: Mg x Ng.
// Requires Mg % 128 == 0, Ng % 64 == 0, Kg == 128 (all call sites satisfy this).
// Block: 256 threads = 8 waves; each wave computes a 16x64 strip with 4
// accumulators. B panel (128x64 = 32 KB) is staged once into LDS in K-pair
// interleaved layout  Bs[(k>>1)*128 + n*2 + (k&1)]  so each WMMA B fragment
// {B[kk][n], B[kk+1][n]} is one aligned ds_load_b64 straight into a VGPR pair
// (no re-pairing movs). 4 independent WMMAs issue per K-step (XDL pipelining;
// only the C-operand accumulation chain carries a dependency).
// flags: bit0 = add bias[col]; bit1 = relu; bit2 = add res[row,col] after act.
__global__ void gemm_wmma_f32(const float* __restrict__ A, const float* __restrict__ Bm,
                              const float* __restrict__ bias, const float* __restrict__ res,
                              float* __restrict__ Cc, int Mg, int Ng, int Kg,
                              long long sA, long long sB, long long sC, int flags) {
    int batch = blockIdx.z;
    A  += (size_t)batch * sA;
    Bm += (size_t)batch * sB;
    Cc += (size_t)batch * sC;
    if (res) res += (size_t)batch * sC;

    __shared__ float Bs[64 * 128];              // 64 K-pairs x (64 cols x 2)

    int stripsN = Ng >> 6;                      // Ng / 64
    int bm = blockIdx.x / stripsN;
    int bn = blockIdx.x % stripsN;

    // ---- cooperative stage of B[:, bn*64 .. bn*64+63], K-pair interleaved ---
    const float* Bsrc = Bm + bn * 64;
    for (int i = threadIdx.x; i < 64 * 16; i += 256) {   // 64 row-pairs x 16 col-quads
        int rp = i >> 4;                         // K-pair index
        int c4 = (i & 15) * 4;                   // column quad
        const float* p0 = Bsrc + (size_t)(2 * rp) * Ng + c4;
        float4 a0 = *(const float4*)p0;          // row 2*rp
        float4 a1 = *(const float4*)(p0 + Ng);   // row 2*rp+1
        float* d = &Bs[rp * 128 + c4 * 2];
        float4 w0 = {a0.x, a1.x, a0.y, a1.y};
        float4 w1 = {a0.z, a1.z, a0.w, a1.w};
        *(float4*)d       = w0;
        *(float4*)(d + 4) = w1;
    }
    __syncthreads();

    int wv   = threadIdx.x >> 5;                // wave 0..7 -> M sub-tile
    int lane = threadIdx.x & 31;
    int half = lane >> 4, l16 = lane & 15;
    int rowBase = bm * 128 + wv * 16;

    const float* arow = A + (size_t)(rowBase + l16) * Kg;  // A frag: (M=l16, K=2*half+v)
    __builtin_prefetch(arow, 0, 3);                        // global_prefetch_b8

    v8f acc0 = {0.f,0.f,0.f,0.f,0.f,0.f,0.f,0.f};
    v8f acc1 = acc0, acc2 = acc0, acc3 = acc0;

#pragma unroll 4
    for (int k0 = 0; k0 < Kg; k0 += 4) {
        int kk = k0 + half * 2;
        v2f a;
        a[0] = arow[kk];
        a[1] = arow[kk + 1];
        // B frag j: elements (K=kk+v, N=16*j+l16) -> adjacent pair in LDS
        const float* bp = &Bs[(kk >> 1) * 128 + l16 * 2];
        v2f b0 = *(const v2f*)(bp);
        v2f b1 = *(const v2f*)(bp + 32);
        v2f b2 = *(const v2f*)(bp + 64);
        v2f b3 = *(const v2f*)(bp + 96);
        acc0 = __builtin_amdgcn_wmma_f32_16x16x4_f32(false, a, false, b0, (short)0, acc0, false, false);
        acc1 = __builtin_amdgcn_wmma_f32_16x16x4_f32(false, a, false, b1, (short)0, acc1, false, false);
        acc2 = __builtin_amdgcn_wmma_f32_16x16x4_f32(false, a, false, b2, (short)0, acc2, false, false);
        acc3 = __builtin_amdgcn_wmma_f32_16x16x4_f32(false, a, false, b3, (short)0, acc3, false, false);
    }

    // ---- epilogue: C frag element (M = v + 8*half, N = 16*j + l16) ----------
    v8f accs[4] = {acc0, acc1, acc2, acc3};
#pragma unroll
    for (int j = 0; j < 4; ++j) {
        int col = bn * 64 + j * 16 + l16;
        float bv = (flags & 1) ? bias[col] : 0.f;
#pragma unroll
        for (int v = 0; v < 8; ++v) {
            int row = rowBase + v + half * 8;
            float val = accs[j][v] + bv;
            if (flags & 2) val = fmaxf(val, 0.f);
            if (flags & 4) val += res[(size_t)row * Ng + col];
            Cc[(size_t)row * Ng + col] = val;
        }
    }
}

// ---------------- misc small kernels ----------------------------------------
__global__ void etype_kernel(const float* __restrict__ ea, int* __restrict__ et) {
    int e = blockIdx.x * 256 + threadIdx.x;
    if (e >= EE) return;
    const float* a = ea + (size_t)e * RR;
    int best = 0; float bv = a[0];
#pragma unroll
    for (int d = 1; d < RR; ++d) { float v = a[d]; if (v > bv) { bv = v; best = d; } }
    et[e] = best;
}

// per-(relation,node) projections qi = <xw_row, q_r>, kj = <xw_row, k_r>; one wave per row
__global__ void qikj_kernel(const float* __restrict__ xw, const float* __restrict__ qw,
                            const float* __restrict__ kw, float* __restrict__ qi,
                            float* __restrict__ kj) {
    int wid  = blockIdx.x * (blockDim.x >> 5) + (threadIdx.x >> 5);
    int lane = threadIdx.x & 31;
    if (wid >= RR * NN) return;
    int r = wid / NN;
    const float* row = xw + (size_t)wid * CC;
    const float* qv = qw + r * CC;
    const float* kv = kw + r * CC;
    float aq = 0.f, ak = 0.f;
    for (int j = lane; j < CC; j += 32) { float xv = row[j]; aq += xv * qv[j]; ak += xv * kv[j]; }
    for (int off = 16; off > 0; off >>= 1) { aq += __shfl_down(aq, off); ak += __shfl_down(ak, off); }
    if (lane == 0) { qi[wid] = aq; kj[wid] = ak; }
}

__global__ void edge_logit_kernel(const int* __restrict__ et, const int* __restrict__ src,
                                  const int* __restrict__ dst, const float* __restrict__ ea,
                                  const float* __restrict__ ce, const float* __restrict__ qi,
                                  const float* __restrict__ kj, float* __restrict__ logit,
                                  unsigned* __restrict__ mEnc) {
    int e = blockIdx.x * 256 + threadIdx.x;
    if (e >= EE) return;
    int r = et[e];
    const float* a  = ea + (size_t)e * RR;
    const float* er = ce + r * RR;
    float ae = 0.f;
#pragma unroll
    for (int d = 0; d < RR; ++d) ae += a[d] * er[d];
    float lg = qi[r * NN + dst[e]] + kj[r * NN + src[e]] + ae;
    lg = (lg >= 0.f) ? lg : 0.2f * lg;        // leaky_relu 0.2
    logit[e] = lg;
    atomicMax(&mEnc[dst[e]], enc_f(lg));
}

__global__ void edge_exp_kernel(const int* __restrict__ dst, const float* __restrict__ logit,
                                const unsigned* __restrict__ mEnc, float* __restrict__ exb,
                                float* __restrict__ den) {
    int e = blockIdx.x * 256 + threadIdx.x;
    if (e >= EE) return;
    int d = dst[e];
    float ex = expf(logit[e] - dec_f(mEnc[d]));
    exb[e] = ex;
    atomicAdd(&den[d], ex);
}

__global__ void edge_msg_kernel(const int* __restrict__ et, const int* __restrict__ src,
                                const int* __restrict__ dst, const float* __restrict__ exb,
                                const float* __restrict__ den, const float* __restrict__ xw,
                                float* __restrict__ agg) {
    int e = blockIdx.x, c = threadIdx.x;
    int d = dst[e];
    float alpha = exb[e] / den[d];
    atomicAdd(&agg[(size_t)d * CC + c], alpha * xw[((size_t)et[e] * NN + src[e]) * CC + c]);
}

__global__ void colstat_kernel(const float* __restrict__ agg, float* __restrict__ colsum,
                               float* __restrict__ colsq) {
    int c = threadIdx.x;                 // 128 threads: one per column
    int r0 = blockIdx.x * 64;
    float s = 0.f, sq = 0.f;
    for (int r = 0; r < 64; ++r) { float v = agg[(size_t)(r0 + r) * CC + c]; s += v; sq += v * v; }
    atomicAdd(&colsum[c], s);
    atomicAdd(&colsq[c], sq);
}

__global__ void bn_relu_kernel(const float* __restrict__ agg, const float* __restrict__ colsum,
                               const float* __restrict__ colsq, const float* __restrict__ g,
                               const float* __restrict__ bb, const float* __restrict__ res,
                               float* __restrict__ xout) {
    size_t idx = (size_t)blockIdx.x * blockDim.x + threadIdx.x;
    if (idx >= (size_t)NN * CC) return;
    int c = (int)(idx & (CC - 1));
    float mean = colsum[c] * (1.f / NN);
    float var  = colsq[c] * (1.f / NN) - mean * mean;
    float val  = (agg[idx] - mean) * rsqrtf(var + EPSV) * g[c] + bb[c];
    if (res) val += res[idx];
    xout[idx] = fmaxf(val, 0.f);
}

__global__ void pool_agg_kernel(const int* __restrict__ src, const int* __restrict__ dst,
                                const float* __restrict__ x, float* __restrict__ agg2) {
    int e = blockIdx.x, c = threadIdx.x;
    atomicAdd(&agg2[(size_t)dst[e] * CC + c], x[(size_t)src[e] * CC + c]);
}

__global__ void score_kernel(const float* __restrict__ agg2, const float* __restrict__ x,
                             const float* __restrict__ pw, const float* __restrict__ pb,
                             float* __restrict__ score) {
    int n = blockIdx.x * 256 + threadIdx.x;
    if (n >= NN) return;
    float s0 = 0.f, s1 = 0.f;
    for (int c = 0; c < CC; ++c) {
        s0 += agg2[(size_t)n * CC + c] * pw[2 * c];
        s1 += x[(size_t)n * CC + c] * pw[2 * c + 1];
    }
    score[n] = tanhf(s0 + pb[0] + s1);
}

// full 1024-element bitonic sort (descending) in LDS; keep top 512 per graph
__global__ void topk_kernel(const float* __restrict__ score, float* __restrict__ topv,
                            int* __restrict__ topi) {
    __shared__ float sc[NPGV];
    __shared__ int   si[NPGV];
    int b = blockIdx.x, t = threadIdx.x;  // 512 threads
    sc[t] = score[b * NPGV + t];           si[t] = t;
    sc[t + 512] = score[b * NPGV + t + 512]; si[t + 512] = t + 512;
    __syncthreads();
    for (int k = 2; k <= NPGV; k <<= 1) {
        for (int j = k >> 1; j > 0; j >>= 1) {
            for (int e = t; e < NPGV; e += 512) {
                int p = e ^ j;
                if (p > e) {
                    bool descend = ((e & k) == 0);
                    float a = sc[e], c = sc[p];
                    if (descend ? (a < c) : (a > c)) {
                        sc[e] = c; sc[p] = a;
                        int ti = si[e]; si[e] = si[p]; si[p] = ti;
                    }
                }
            }
            __syncthreads();
        }
    }
    topv[b * KPOOLV + t] = sc[t];
    topi[b * KPOOLV + t] = si[t];
}

__global__ void gather_kernel(const float* __restrict__ x, const int* __restrict__ topi,
                              const float* __restrict__ topv, float* __restrict__ xp) {
    int bk = blockIdx.x, c = threadIdx.x;
    int b = bk >> 9;
    int node = b * NPGV + topi[bk];
    xp[(size_t)bk * CC + c] = x[(size_t)node * CC + c] * topv[bk];
}

// one block per (b, h, q-row); K/V stream from L2 (64KB/head, fully resident)
__global__ void attn_kernel(const float* __restrict__ q, const float* __restrict__ k,
                            const float* __restrict__ v, float* __restrict__ o,
                            int Tq, int Tk, float scale) {
    int b = blockIdx.z, h = blockIdx.y, qr = blockIdx.x;
    int tid = threadIdx.x;                // 256 threads
    __shared__ float qs[DH];
    __shared__ float sc[512];
    __shared__ float red[256];
    __shared__ float part[8][DH];
    const float* qrow = q + ((size_t)(b * Tq + qr)) * CC + h * DH;
    if (tid < DH) qs[tid] = qrow[tid];
    __syncthreads();
    for (int t = tid; t < Tk; t += 256) {
        const float* krow = k + ((size_t)(b * Tk + t)) * CC + h * DH;
        float s = 0.f;
#pragma unroll
        for (int d = 0; d < DH; ++d) s += qs[d] * krow[d];
        sc[t] = s * scale;
    }
    __syncthreads();
    float m = -3.0e38f;
    for (int t = tid; t < Tk; t += 256) m = fmaxf(m, sc[t]);
    red[tid] = m; __syncthreads();
    for (int s2 = 128; s2 > 0; s2 >>= 1) {
        if (tid < s2) red[tid] = fmaxf(red[tid], red[tid + s2]);
        __syncthreads();
    }
    float mx = red[0]; __syncthreads();
    float sum = 0.f;
    for (int t = tid; t < Tk; t += 256) { float e = expf(sc[t] - mx); sc[t] = e; sum += e; }
    red[tid] = sum; __syncthreads();
    for (int s2 = 128; s2 > 0; s2 >>= 1) {
        if (tid < s2) red[tid] += red[tid + s2];
        __syncthreads();
    }
    float den = red[0];
    int d = tid & (DH - 1), chunk = tid >> 5;
    float acc = 0.f;
    for (int t = chunk; t < Tk; t += 8)
        acc += sc[t] * v[((size_t)(b * Tk + t)) * CC + h * DH + d];
    part[chunk][d] = acc; __syncthreads();
    if (tid < DH) {
        float s = 0.f;
#pragma unroll
        for (int c = 0; c < 8; ++c) s += part[c][tid];
        o[((size_t)(b * Tq + qr)) * CC + h * DH + tid] = s / den;
    }
}

__global__ void s0_kernel(const float* __restrict__ seed, float* __restrict__ s0) {
    int idx = blockIdx.x * 256 + threadIdx.x;
    if (idx >= BB * NSEEDS * CC) return;
    int c = idx & (CC - 1), s = (idx >> 7) & (NSEEDS - 1);
    s0[idx] = seed[s * CC + c];
}

__global__ void batchout_kernel(float* __restrict__ outp) {
    int i = blockIdx.x * 256 + threadIdx.x;
    if (i >= BB * KPOOLV) return;
    outp[i] = (float)(i >> 9);            // repeat(arange(B), KPOOL), as float
}

// ---------------- host orchestration ----------------------------------------
static void launch_gemm(const float* A, const float* Bm, const float* bias, const float* res,
                        float* Cc, int Mg, int Ng, int Kg, int batch,
                        long long sA, long long sB, long long sC, int flags, hipStream_t st) {
    int blocks = (Mg / 128) * (Ng / 64);
    gemm_wmma_f32<<<dim3(blocks, 1, batch), dim3(256), 0, st>>>(A, Bm, bias, res, Cc,
                                                               Mg, Ng, Kg, sA, sB, sC, flags);
}

extern "C" void kernel_launch(void* const* d_in, const int* in_sizes, int n_in,
                              void* d_out, int out_size, void* d_ws, size_t ws_size,
                              hipStream_t stream) {
    (void)in_sizes; (void)n_in; (void)out_size; (void)ws_size;
    const float* x0        = (const float*)d_in[0];
    const float* edge_attr = (const float*)d_in[1];
    const int*   eidx      = (const int*)d_in[2];
    const int*   src = eidx, *dst = eidx + EE;
    const float* conv_w    = (const float*)d_in[4];
    const float* conv_q    = (const float*)d_in[5];
    const float* conv_k    = (const float*)d_in[6];
    const float* conv_e    = (const float*)d_in[7];
    const float* bn_g      = (const float*)d_in[9];
    const float* bn_b      = (const float*)d_in[10];
    const float* pool_w    = (const float*)d_in[11];
    const float* pool_b    = (const float*)d_in[12];
    const float* mab_w     = (const float*)d_in[13];
    const float* mab_b     = (const float*)d_in[14];
    const float* mab_lw    = (const float*)d_in[15];
    const float* mab_lb    = (const float*)d_in[16];
    const float* pma_lw    = (const float*)d_in[17];
    const float* pma_lb    = (const float*)d_in[18];
    const float* seed      = (const float*)d_in[19];
    float* out = (float*)d_out;

    char* base = (char*)d_ws;
    size_t off = 0;
    auto alloc = [&](size_t bytes) -> void* {
        off = (off + 255) & ~(size_t)255;
        void* p = base + off;
        off += bytes;
        return p;
    };
    int*      etype  = (int*)     alloc((size_t)EE * 4);
    float*    xw     = (float*)   alloc((size_t)RR * NN * CC * 4);
    float*    qi     = (float*)   alloc((size_t)RR * NN * 4);
    float*    kj     = (float*)   alloc((size_t)RR * NN * 4);
    float*    logit  = (float*)   alloc((size_t)EE * 4);
    float*    exb    = (float*)   alloc((size_t)EE * 4);
    unsigned* mEnc   = (unsigned*)alloc((size_t)NN * 4);
    float*    den    = (float*)   alloc((size_t)NN * 4);
    float*    agg    = (float*)   alloc((size_t)NN * CC * 4);   // reused for pool agg2
    float*    xA     = (float*)   alloc((size_t)NN * CC * 4);
    float*    xB     = (float*)   alloc((size_t)NN * CC * 4);
    float*    colsum = (float*)   alloc(CC * 4);
    float*    colsq  = (float*)   alloc(CC * 4);
    float*    score  = (float*)   alloc((size_t)NN * 4);
    float*    topv   = (float*)   alloc((size_t)BB * KPOOLV * 4);
    int*      topi   = (int*)     alloc((size_t)BB * KPOOLV * 4);
    float*    xp     = (float*)   alloc((size_t)BB * KPOOLV * CC * 4);
    float*    qb     = (float*)   alloc((size_t)BB * KPOOLV * CC * 4);
    float*    kb     = (float*)   alloc((size_t)BB * KPOOLV * CC * 4);
    float*    vb     = (float*)   alloc((size_t)BB * KPOOLV * CC * 4);
    float*    ob     = (float*)   alloc((size_t)BB * KPOOLV * CC * 4);
    float*    hb     = (float*)   alloc((size_t)BB * KPOOLV * CC * 4);
    float*    encb   = (float*)   alloc((size_t)BB * KPOOLV * CC * 4);
    float*    pinb   = (float*)   alloc((size_t)BB * KPOOLV * CC * 4);
    float*    s0b    = (float*)   alloc((size_t)BB * NSEEDS * CC * 4);

    etype_kernel<<<EE / 256, 256, 0, stream>>>(edge_attr, etype);

    const float* xin = x0;
    float* xouts[2] = {xA, xB};
    for (int l = 0; l < 2; ++l) {
        // xw[r] = x @ W[l][r]   (batched over R via blockIdx.z)
        launch_gemm(xin, conv_w + (size_t)l * RR * CC * CC, nullptr, nullptr, xw,
                    NN, CC, CC, RR, 0, (long long)CC * CC, (long long)NN * CC, 0, stream);
        qikj_kernel<<<(RR * NN) / 4, 128, 0, stream>>>(xw, conv_q + l * RR * CC,
                                                       conv_k + l * RR * CC, qi, kj);
        hipMemsetAsync(mEnc, 0, (size_t)NN * 4, stream);   // encoded -inf
        hipMemsetAsync(den, 0, (size_t)NN * 4, stream);
        hipMemsetAsync(agg, 0, (size_t)NN * CC * 4, stream);
        hipMemsetAsync(colsum, 0, CC * 4, stream);
        hipMemsetAsync(colsq, 0, CC * 4, stream);
        edge_logit_kernel<<<EE / 256, 256, 0, stream>>>(etype, src, dst, edge_attr,
                                                        conv_e + l * RR * RR, qi, kj, logit, mEnc);
        edge_exp_kernel<<<EE / 256, 256, 0, stream>>>(dst, logit, mEnc, exb, den);
        edge_msg_kernel<<<EE, CC, 0, stream>>>(etype, src, dst, exb, den, xw, agg);
        colstat_kernel<<<NN / 64, CC, 0, stream>>>(agg, colsum, colsq);
        bn_relu_kernel<<<(NN * CC) / 256, 256, 0, stream>>>(agg, colsum, colsq,
                                                            bn_g + l * CC, bn_b + l * CC,
                                                            (l > 0) ? xin : nullptr, xouts[l]);
        xin = xouts[l];
    }
    const float* xf = xB;

    // SAGPooling
    hipMemsetAsync(agg, 0, (size_t)NN * CC * 4, stream);
    pool_agg_kernel<<<EE, CC, 0, stream>>>(src, dst, xf, agg);
    score_kernel<<<NN / 256, 256, 0, stream>>>(agg, xf, pool_w, pool_b, score);
    topk_kernel<<<BB, 512, 0, stream>>>(score, topv, topi);
    gather_kernel<<<BB * KPOOLV, CC, 0, stream>>>(xf, topi, topv, xp);

    const float scale = 0.1767766953f;    // 1/sqrt(32)
    const int MK = BB * KPOOLV;           // 16384
    const int MS = BB * NSEEDS;           // 128
    auto W = [&](int i, int j) { return mab_w + ((size_t)(i * 4 + j)) * CC * CC; };
    auto Bv = [&](int i, int j) { return mab_b + (size_t)(i * 4 + j) * CC; };

    // encoder SAB: enc = MAB(xp, xp)
    launch_gemm(xp, W(0, 0), Bv(0, 0), nullptr, qb, MK, CC, CC, 1, 0, 0, 0, 1, stream);
    launch_gemm(xp, W(0, 1), Bv(0, 1), nullptr, kb, MK, CC, CC, 1, 0, 0, 0, 1, stream);
    launch_gemm(xp, W(0, 2), Bv(0, 2), nullptr, vb, MK, CC, CC, 1, 0, 0, 0, 1, stream);
    attn_kernel<<<dim3(KPOOLV, NHEADS, BB), 256, 0, stream>>>(qb, kb, vb, ob, KPOOLV, KPOOLV, scale);
    launch_gemm(ob, W(0, 3), Bv(0, 3), xp, hb, MK, CC, CC, 1, 0, 0, 0, 5, stream);          // h = xp + o@w3+b3
    launch_gemm(hb, mab_lw + 0 * CC * CC, mab_lb + 0 * CC, hb, encb, MK, CC, CC, 1, 0, 0, 0, 7, stream);
    launch_gemm(encb, pma_lw, pma_lb, nullptr, pinb, MK, CC, CC, 1, 0, 0, 0, 3, stream);    // pin = relu(..)

    // PMA: p = MAB(s0, pin)
    s0_kernel<<<(BB * NSEEDS * CC + 255) / 256, 256, 0, stream>>>(seed, s0b);
    launch_gemm(s0b, W(1, 0), Bv(1, 0), nullptr, qb, MS, CC, CC, 1, 0, 0, 0, 1, stream);
    launch_gemm(pinb, W(1, 1), Bv(1, 1), nullptr, kb, MK, CC, CC, 1, 0, 0, 0, 1, stream);
    launch_gemm(pinb, W(1, 2), Bv(1, 2), nullptr, vb, MK, CC, CC, 1, 0, 0, 0, 1, stream);
    attn_kernel<<<dim3(NSEEDS, NHEADS, BB), 256, 0, stream>>>(qb, kb, vb, ob, NSEEDS, KPOOLV, scale);
    launch_gemm(ob, W(1, 3), Bv(1, 3), s0b, hb, MS, CC, CC, 1, 0, 0, 0, 5, stream);
    launch_gemm(hb, mab_lw + 1 * CC * CC, mab_lb + 1 * CC, hb, encb, MS, CC, CC, 1, 0, 0, 0, 7, stream); // p

    // decoder SAB: dec = MAB(p, p) -> d_out
    launch_gemm(encb, W(2, 0), Bv(2, 0), nullptr, qb, MS, CC, CC, 1, 0, 0, 0, 1, stream);
    launch_gemm(encb, W(2, 1), Bv(2, 1), nullptr, kb, MS, CC, CC, 1, 0, 0, 0, 1, stream);
    launch_gemm(encb, W(2, 2), Bv(2, 2), nullptr, vb, MS, CC, CC, 1, 0, 0, 0, 1, stream);
    attn_kernel<<<dim3(NSEEDS, NHEADS, BB), 256, 0, stream>>>(qb, kb, vb, ob, NSEEDS, NSEEDS, scale);
    launch_gemm(ob, W(2, 3), Bv(2, 3), encb, hb, MS, CC, CC, 1, 0, 0, 0, 5, stream);
    launch_gemm(hb, mab_lw + 2 * CC * CC, mab_lb + 2 * CC, hb, out, MS, CC, CC, 1, 0, 0, 0, 7, stream);

    batchout_kernel<<<(BB * KPOOLV + 255) / 256, 256, 0, stream>>>(out + BB * NSEEDS * CC);
}